// PointSeg_75857712382171
// MI455X (gfx1250) — compile-verified
//
#include <hip/hip_runtime.h>
#include <hip/hip_bf16.h>

// ---------------------------------------------------------------------------
// PointNet++ part-seg forward, CDNA5 (gfx1250, wave32).
// Conv layers run on V_WMMA_F32_16X16X4_F32 with a guard-free inner loop:
// weights are zero-padded to (Mp x Kp) in workspace, inputs always carry
// Kp (=ceil4) channel rows, so the K-loop is branchless and pipelineable.
// ---------------------------------------------------------------------------

typedef __attribute__((ext_vector_type(2))) float v2f;
typedef __attribute__((ext_vector_type(8))) float v8f;

#define NB 16          // batch
#define GWAVES 4       // waves per GEMM block; each wave does a 16x32 tile

static inline int ceil_div(int a, int b) { return (a + b - 1) / b; }
static inline int gs_blocks(size_t total) {
    size_t b = (total + 255) / 256;
    return (int)(b < 4096 ? b : 4096);
}

// ---------------------------------------------------------------------------
// Zero-pad weights: Wp (Mp x Kp) <- W (M x K), pad = 0
// ---------------------------------------------------------------------------
__global__ void k_padw(const float* __restrict__ W, float* __restrict__ Wp,
                       int M, int K, int Mp, int Kp)
{
    size_t total = (size_t)Mp * Kp;
    for (size_t i = blockIdx.x * (size_t)blockDim.x + threadIdx.x; i < total;
         i += (size_t)gridDim.x * blockDim.x) {
        int r = (int)(i / (size_t)Kp), c = (int)(i % (size_t)Kp);
        Wp[i] = (r < M && c < K) ? W[(size_t)r * K + c] : 0.f;
    }
}

__global__ void k_zero(float* __restrict__ p, size_t n)
{
    for (size_t i = blockIdx.x * (size_t)blockDim.x + threadIdx.x; i < n;
         i += (size_t)gridDim.x * blockDim.x)
        p[i] = 0.f;
}

// ---------------------------------------------------------------------------
// WMMA fp32 GEMM:  Y[b](M x P) = Wp(Mp x Kp) * X[b](Kp x P)  (+bias, +relu)
// Wave computes a 16x32 tile (2 accs share the A fragment). Kp % 4 == 0,
// P % 32 == 0, X batch stride = Kp*P, Y batch stride = Ys*P, stores row < M.
// A frag (16x4 fp32): lanes 0-15 -> K {0,1}, lanes 16-31 -> K {2,3}.
// D: VGPR r -> row r (lanes 0-15) / row 8+r (lanes 16-31), col = lane&15.
// ---------------------------------------------------------------------------
__global__ void k_gemm(const float* __restrict__ Wp, const float* __restrict__ X,
                       float* __restrict__ Y, const float* __restrict__ bias,
                       int M, int Kp, int P, int relu, int Ys)
{
    int wave = threadIdx.x >> 5;
    int lane = threadIdx.x & 31;
    int m0 = blockIdx.x * 16;
    int p0 = (blockIdx.y * GWAVES + wave) * 32;
    if (p0 >= P) return;                 // wave-uniform: EXEC all-ones at WMMA
    int b = blockIdx.z;

    int half = lane >> 4;                // 0 or 1
    int l16  = lane & 15;
    int kb   = half * 2;                 // this half-wave's K offsets

    const float* wrow = Wp + (size_t)(m0 + l16) * Kp + kb;           // aligned 8B
    const float* xc   = X + (size_t)b * Kp * P + (size_t)kb * P + p0 + l16;
    float*       Yb   = Y + (size_t)b * Ys * P;

    v8f acc0 = {0.f, 0.f, 0.f, 0.f, 0.f, 0.f, 0.f, 0.f};
    v8f acc1 = {0.f, 0.f, 0.f, 0.f, 0.f, 0.f, 0.f, 0.f};

    for (int kk = 0; kk < Kp; kk += 4) {
        v2f a = *(const v2f*)(wrow);     // W[row][kk+kb], W[row][kk+kb+1]
        v2f b0, b1;
        b0.x = xc[0];
        b0.y = xc[(size_t)P];
        b1.x = xc[16];
        b1.y = xc[(size_t)P + 16];
        acc0 = __builtin_amdgcn_wmma_f32_16x16x4_f32(
                   false, a, false, b0, (short)0, acc0, false, false);
        acc1 = __builtin_amdgcn_wmma_f32_16x16x4_f32(
                   false, a, false, b1, (short)0, acc1, false, false);
        wrow += 4;
        xc   += (size_t)4 * P;
    }

#pragma unroll
    for (int r = 0; r < 8; ++r) {
        int row = m0 + r + 8 * half;
        if (row < M) {
            float v0 = acc0[r], v1 = acc1[r];
            if (bias) { v0 += bias[row]; v1 += bias[row]; }
            if (relu) { v0 = fmaxf(v0, 0.f); v1 = fmaxf(v1, 0.f); }
            Yb[(size_t)row * P + p0 + l16]      = v0;
            Yb[(size_t)row * P + p0 + 16 + l16] = v1;
        }
    }
}

// ---------------------------------------------------------------------------
// BatchNorm (training-mode, gamma=1, beta=0): per-channel mean/var over (B,P)
// ---------------------------------------------------------------------------
__global__ void k_bn_stats(const float* __restrict__ X, float* __restrict__ mean,
                           float* __restrict__ var, int C, int P, int Bn)
{
    __shared__ float ssum[256], ssq[256];
    int c = blockIdx.x, tid = threadIdx.x;
    float s = 0.f, q = 0.f;
    for (int b = 0; b < Bn; ++b) {
        const float* x = X + ((size_t)b * C + c) * P;
        for (int i = tid; i < P; i += blockDim.x) {
            float v = x[i]; s += v; q += v * v;
        }
    }
    ssum[tid] = s; ssq[tid] = q;
    __syncthreads();
    for (int st = blockDim.x / 2; st > 0; st >>= 1) {
        if (tid < st) { ssum[tid] += ssum[tid + st]; ssq[tid] += ssq[tid + st]; }
        __syncthreads();
    }
    if (tid == 0) {
        float n = (float)Bn * (float)P;
        float m = ssum[0] / n;
        mean[c] = m;
        var[c]  = fmaxf(ssq[0] / n - m * m, 0.f);
    }
}

__global__ void k_bn_apply(float* __restrict__ X, const float* __restrict__ mean,
                           const float* __restrict__ var, int C, int P, int Bn)
{
    size_t total = (size_t)Bn * C * P;
    for (size_t i = blockIdx.x * (size_t)blockDim.x + threadIdx.x; i < total;
         i += (size_t)gridDim.x * blockDim.x) {
        int c = (int)((i / (size_t)P) % (size_t)C);
        float v = (X[i] - mean[c]) * rsqrtf(var[c] + 1e-5f);
        X[i] = fmaxf(v, 0.f);
    }
}

// ---------------------------------------------------------------------------
// Farthest point sampling: one block per batch element, sequential scan.
// Matches lax.scan semantics: emits `far` before update; first index is 0.
// ---------------------------------------------------------------------------
__global__ void k_fps(const float* __restrict__ xyz, int* __restrict__ fidx,
                      int N, int S)
{
    __shared__ float sd[2048];
    __shared__ float rv[256];
    __shared__ int   ri[256];
    int b = blockIdx.x, tid = threadIdx.x;
    const float* px = xyz + (size_t)b * N * 3;
    for (int i = tid; i < N; i += blockDim.x) sd[i] = 3.4e38f;
    __syncthreads();
    int far = 0;
    for (int it = 0; it < S; ++it) {
        if (tid == 0) fidx[b * S + it] = far;
        float fx = px[far * 3 + 0], fy = px[far * 3 + 1], fz = px[far * 3 + 2];
        float bv = -1.f; int bi = 0;
        for (int i = tid; i < N; i += blockDim.x) {
            float dx = px[i * 3 + 0] - fx;
            float dy = px[i * 3 + 1] - fy;
            float dz = px[i * 3 + 2] - fz;
            float nd = fminf(sd[i], dx * dx + dy * dy + dz * dz);
            sd[i] = nd;
            if (nd > bv) { bv = nd; bi = i; }     // ascending i: first-max wins
        }
        rv[tid] = bv; ri[tid] = bi;
        __syncthreads();
        for (int st = blockDim.x / 2; st > 0; st >>= 1) {
            if (tid < st) {
                if (rv[tid + st] > rv[tid] ||
                    (rv[tid + st] == rv[tid] && ri[tid + st] < ri[tid])) {
                    rv[tid] = rv[tid + st]; ri[tid] = ri[tid + st];
                }
            }
            __syncthreads();
        }
        far = ri[0];
        __syncthreads();
    }
}

__global__ void k_gather_xyz(const float* __restrict__ xyz, const int* __restrict__ fidx,
                             float* __restrict__ nxyz, int N, int S)
{
    int i = blockIdx.x * blockDim.x + threadIdx.x;
    int total = NB * S * 3;
    if (i >= total) return;
    int c = i % 3, s = (i / 3) % S, b = i / (3 * S);
    nxyz[i] = xyz[((size_t)b * N + fidx[b * S + s]) * 3 + c];
}

// Ball query: ascending-index scan == reference sort + first-hit fill.
__global__ void k_ball(const float* __restrict__ xyz, const float* __restrict__ nxyz,
                       int* __restrict__ gidx, int N, int S, int ns, float r2)
{
    int g = blockIdx.x * blockDim.x + threadIdx.x;
    if (g >= NB * S) return;
    int b = g / S, s = g % S;
    const float* px = xyz + (size_t)b * N * 3;
    float cx = nxyz[(b * S + s) * 3 + 0];
    float cy = nxyz[(b * S + s) * 3 + 1];
    float cz = nxyz[(b * S + s) * 3 + 2];
    int* out = gidx + ((size_t)b * S + s) * ns;
    int cnt = 0, first = 0;
    for (int j = 0; j < N && cnt < ns; ++j) {
        float dx = px[j * 3 + 0] - cx;
        float dy = px[j * 3 + 1] - cy;
        float dz = px[j * 3 + 2] - cz;
        if (dx * dx + dy * dy + dz * dz < r2) {
            if (cnt == 0) first = j;
            out[cnt++] = j;
        }
    }
    for (int k = cnt; k < ns; ++k) out[k] = first;
}

// Grouped tensor (B, Cpad, S, ns): rel-xyz + gathered feats, pad channels = 0.
__global__ void k_group(const float* __restrict__ xyz, const float* __restrict__ nxyz,
                        const float* __restrict__ feats, const int* __restrict__ gidx,
                        float* __restrict__ g, int N, int S, int ns, int Cf, int Cpad)
{
    int C = 3 + Cf;
    size_t total = (size_t)NB * Cpad * S * ns;
    for (size_t i = blockIdx.x * (size_t)blockDim.x + threadIdx.x; i < total;
         i += (size_t)gridDim.x * blockDim.x) {
        int n = (int)(i % (size_t)ns);
        int s = (int)((i / (size_t)ns) % (size_t)S);
        int c = (int)((i / ((size_t)ns * S)) % (size_t)Cpad);
        int b = (int)(i / ((size_t)ns * S * Cpad));
        float v = 0.f;
        if (c < C) {
            int j = gidx[((size_t)b * S + s) * ns + n];
            if (c < 3)
                v = xyz[((size_t)b * N + j) * 3 + c] - nxyz[((size_t)b * S + s) * 3 + c];
            else
                v = feats[((size_t)b * Cf + (c - 3)) * N + j];
        }
        g[i] = v;
    }
}

__global__ void k_maxns(const float* __restrict__ g, float* __restrict__ out,
                        int C, int S, int ns)
{
    size_t total = (size_t)NB * C * S;
    for (size_t i = blockIdx.x * (size_t)blockDim.x + threadIdx.x; i < total;
         i += (size_t)gridDim.x * blockDim.x) {
        const float* p = g + i * (size_t)ns;
        float m = p[0];
        for (int n = 1; n < ns; ++n) m = fmaxf(m, p[n]);
        out[i] = m;
    }
}

// 3-NN with stable tie order (strict <) + inverse-distance weights.
__global__ void k_three_nn(const float* __restrict__ unk, const float* __restrict__ kn,
                           int* __restrict__ idx3, float* __restrict__ w3,
                           int Nu, int Nk)
{
    int g = blockIdx.x * blockDim.x + threadIdx.x;
    if (g >= NB * Nu) return;
    int b = g / Nu, n = g % Nu;
    float ux = unk[((size_t)b * Nu + n) * 3 + 0];
    float uy = unk[((size_t)b * Nu + n) * 3 + 1];
    float uz = unk[((size_t)b * Nu + n) * 3 + 2];
    const float* pk = kn + (size_t)b * Nk * 3;
    float d0 = 3.4e38f, d1 = 3.4e38f, d2 = 3.4e38f;
    int i0 = 0, i1 = 0, i2 = 0;
    for (int j = 0; j < Nk; ++j) {
        float dx = pk[j * 3 + 0] - ux;
        float dy = pk[j * 3 + 1] - uy;
        float dz = pk[j * 3 + 2] - uz;
        float d = dx * dx + dy * dy + dz * dz;
        if (d < d0)      { d2 = d1; i2 = i1; d1 = d0; i1 = i0; d0 = d; i0 = j; }
        else if (d < d1) { d2 = d1; i2 = i1; d1 = d; i1 = j; }
        else if (d < d2) { d2 = d; i2 = j; }
    }
    float w0 = 1.f / (d0 + 1e-8f), w1 = 1.f / (d1 + 1e-8f), w2 = 1.f / (d2 + 1e-8f);
    float ws = w0 + w1 + w2;
    idx3[(size_t)g * 3 + 0] = i0; idx3[(size_t)g * 3 + 1] = i1; idx3[(size_t)g * 3 + 2] = i2;
    w3[(size_t)g * 3 + 0] = w0 / ws; w3[(size_t)g * 3 + 1] = w1 / ws; w3[(size_t)g * 3 + 2] = w2 / ws;
}

// Interpolate known feats to unknown points; concat skip feats (channels Ck..Ck+Cu).
__global__ void k_interp(const float* __restrict__ knf, const int* __restrict__ idx3,
                         const float* __restrict__ w3, const float* __restrict__ skip,
                         float* __restrict__ out, int Ck, int Cu, int Nu, int Nk)
{
    int C = Ck + Cu;
    size_t total = (size_t)NB * C * Nu;
    for (size_t i = blockIdx.x * (size_t)blockDim.x + threadIdx.x; i < total;
         i += (size_t)gridDim.x * blockDim.x) {
        int n = (int)(i % (size_t)Nu);
        int c = (int)((i / (size_t)Nu) % (size_t)C);
        int b = (int)(i / ((size_t)Nu * C));
        float v;
        if (c < Ck) {
            const int*   id = idx3 + ((size_t)b * Nu + n) * 3;
            const float* w  = w3   + ((size_t)b * Nu + n) * 3;
            const float* f  = knf  + ((size_t)b * Ck + c) * Nk;
            v = f[id[0]] * w[0] + f[id[1]] * w[1] + f[id[2]] * w[2];
        } else {
            v = skip[((size_t)b * Cu + (c - Ck)) * Nu + n];
        }
        out[i] = v;
    }
}

// pred_s: mean over halves of x (B,3,2048) -> (B,6) in (c,h) order
__global__ void k_pred_s(const float* __restrict__ x, float* __restrict__ out)
{
    int g = blockIdx.x * blockDim.x + threadIdx.x;
    if (g >= NB * 6) return;
    int b = g / 6, c = (g % 6) / 2, h = g % 2;
    const float* p = x + ((size_t)b * 3 + c) * 2048 + h * 1024;
    float s = 0.f;
    for (int i = 0; i < 1024; ++i) s += p[i];
    out[g] = s * (1.f / 1024.f);
}

// softmax over 2 channels; segc has Cs-row batch stride; writes (B,N,2)
__global__ void k_softmax2(const float* __restrict__ segc, float* __restrict__ out,
                           int N, int Cs)
{
    int g = blockIdx.x * blockDim.x + threadIdx.x;
    if (g >= NB * N) return;
    int b = g / N, n = g % N;
    float a = segc[((size_t)b * Cs + 0) * N + n];
    float c = segc[((size_t)b * Cs + 1) * N + n];
    float m = fmaxf(a, c);
    float e0 = expf(a - m), e1 = expf(c - m);
    float inv = 1.f / (e0 + e1);
    out[(size_t)g * 2 + 0] = e0 * inv;
    out[(size_t)g * 2 + 1] = e1 * inv;
}

// cls_feat: out (B, Np, C) = f(B, C, N)[:, :, perm] transposed
__global__ void k_gather_cls(const float* __restrict__ f, const int* __restrict__ perm,
                             float* __restrict__ out, int N, int Np, int C)
{
    size_t total = (size_t)NB * Np * C;
    for (size_t i = blockIdx.x * (size_t)blockDim.x + threadIdx.x; i < total;
         i += (size_t)gridDim.x * blockDim.x) {
        int c = (int)(i % (size_t)C);
        int p = (int)((i / (size_t)C) % (size_t)Np);
        int b = (int)(i / ((size_t)C * Np));
        out[i] = f[((size_t)b * C + c) * N + perm[p]];
    }
}

// ---------------------------------------------------------------------------
// Host orchestration
// ---------------------------------------------------------------------------
extern "C" void kernel_launch(void* const* d_in, const int* in_sizes, int n_in,
                              void* d_out, int out_size, void* d_ws, size_t ws_size,
                              hipStream_t stream)
{
    (void)in_sizes; (void)n_in; (void)out_size; (void)ws_size;

    // Input flattening assumption (recursive dict-insertion order):
    // 0: xyz | 1..36: sa[i][j] (W,g,b) | 37..63: fp | 64..66: fc |
    // 67..74: iou (W,g,b)x2 + (W,bias) | 75..76: seg | 77..82: feat | 83: perm
    const float* xyz = (const float*)d_in[0];
    auto F = [&](int i) { return (const float*)d_in[i]; };
    const int* perm = (const int*)d_in[83];

    // ---- workspace layout (floats) ----
    float* ws   = (float*)d_ws;
    float* pA   = ws;                                  // 16*68*1024*32 = 35,651,584
    float* pB   = ws + 35651584UL;                     // 35,651,584
    float* xyz1 = ws + 71303168UL;                     // 16*1024*3
    float* xyz2 = xyz1 + 49152;                        // 16*256*3
    float* xyz3 = xyz2 + 12288;                        // 16*64*3
    float* xyz4 = xyz3 + 3072;                         // 16*16*3
    float* f1   = xyz4 + 768;                          // up to 16*128*1024
    float* f2   = f1 + 2097152;                        // up to 16*256*256
    float* f3   = f2 + 1048576;                        // 16*256*64
    float* f4   = f3 + 262144;                         // 16*512*16
    float* f0   = f4 + 131072;                         // 16*128*2048
    float* dbuf = f0 + 4194304;                        // 16*128*2048
    float* wpad = dbuf + 4194304;                      // max Mp*Kp = 262,144
    float* w3b  = wpad + 262144;                       // 16*2048*3
    float* meanb = w3b + 98304;                        // 512
    float* varb  = meanb + 512;                        // 512
    int* fidx = (int*)(varb + 512);                    // 16*1024
    int* gidx = fidx + 16384;                          // 16*1024*32
    int* idx3 = gidx + 524288;                         // 16*2048*3

    // conv: X must have Kp=ceil4(K) channel rows per batch (pad rows zero);
    // Y gets Ys-row batch stride, rows [0,M) written.
    auto conv = [&](const float* Wm, int M, int K, const float* X, float* Y, int P,
                    const float* bias, int relu, int Ys) {
        int Mp = ceil_div(M, 16) * 16;
        int Kp = ceil_div(K, 4) * 4;
        k_padw<<<gs_blocks((size_t)Mp * Kp), 256, 0, stream>>>(Wm, wpad, M, K, Mp, Kp);
        dim3 grid(Mp / 16, ceil_div(P, 32 * GWAVES), NB);
        k_gemm<<<grid, GWAVES * 32, 0, stream>>>(wpad, X, Y, bias, M, Kp, P, relu, Ys);
    };
    auto conv_bn = [&](const float* Wm, int M, int K, const float* X, float* Y, int P) {
        conv(Wm, M, K, X, Y, P, nullptr, 0, M);
        k_bn_stats<<<M, 256, 0, stream>>>(Y, meanb, varb, M, P, NB);
        k_bn_apply<<<gs_blocks((size_t)NB * M * P), 256, 0, stream>>>(Y, meanb, varb, M, P, NB);
    };

    // ---- set abstraction stage ----
    auto sa_stage = [&](const float* xin, const float* fin, int Cf, int N, int S, float radius,
                        const float* W0, int C1, const float* W1, int C2,
                        const float* W2, int C3, float* xout, float* fout) {
        k_fps<<<NB, 256, 0, stream>>>(xin, fidx, N, S);
        k_gather_xyz<<<ceil_div(NB * S * 3, 256), 256, 0, stream>>>(xin, fidx, xout, N, S);
        k_ball<<<ceil_div(NB * S, 256), 256, 0, stream>>>(xin, xout, gidx, N, S, 32,
                                                          radius * radius);
        int C0 = 3 + Cf;
        int C0p = ceil_div(C0, 4) * 4;
        k_group<<<gs_blocks((size_t)NB * C0p * S * 32), 256, 0, stream>>>(
            xin, xout, fin, gidx, pA, N, S, 32, Cf, C0p);
        int P = S * 32;
        conv_bn(W0, C1, C0, pA, pB, P);
        conv_bn(W1, C2, C1, pB, pA, P);
        conv_bn(W2, C3, C2, pA, pB, P);
        k_maxns<<<gs_blocks((size_t)NB * C3 * S), 256, 0, stream>>>(pB, fout, C3, S, 32);
    };

    sa_stage(xyz,  nullptr, 0,   2048, 1024, 0.1f, F(1),  32,  F(4),  32,  F(7),  64,  xyz1, f1);
    sa_stage(xyz1, f1,      64,  1024, 256,  0.2f, F(10), 64,  F(13), 64,  F(16), 128, xyz2, f2);
    sa_stage(xyz2, f2,      128, 256,  64,   0.4f, F(19), 128, F(22), 128, F(25), 256, xyz3, f3);
    sa_stage(xyz3, f3,      256, 64,   16,   0.8f, F(28), 256, F(31), 256, F(34), 512, xyz4, f4);

    // ---- feature propagation stage (Ck+Cu always multiple of 4) ----
    auto fp_stage = [&](const float* uxyz, const float* kxyz, const float* skip, int Cu,
                        const float* knf, int Ck, int Nu, int Nk,
                        const float* W0, int C1, const float* W1, int C2,
                        const float* W2, int C3, float* fout) {
        k_three_nn<<<ceil_div(NB * Nu, 256), 256, 0, stream>>>(uxyz, kxyz, idx3, w3b, Nu, Nk);
        k_interp<<<gs_blocks((size_t)NB * (Ck + Cu) * Nu), 256, 0, stream>>>(
            knf, idx3, w3b, skip, pA, Ck, Cu, Nu, Nk);
        if (W2) {
            conv_bn(W0, C1, Ck + Cu, pA, pB, Nu);
            conv_bn(W1, C2, C1, pB, pA, Nu);
            conv_bn(W2, C3, C2, pA, fout, Nu);
        } else {
            conv_bn(W0, C1, Ck + Cu, pA, pB, Nu);
            conv_bn(W1, C2, C1, pB, fout, Nu);
        }
    };

    fp_stage(xyz3, xyz4, f3, 256, f4, 512, 64,   16,   F(58), 256, F(61), 256, nullptr, 0, f3);
    fp_stage(xyz2, xyz3, f2, 128, f3, 256, 256,  64,   F(52), 256, F(55), 256, nullptr, 0, f2);
    fp_stage(xyz1, xyz2, f1, 64,  f2, 256, 1024, 256,  F(46), 256, F(49), 128, nullptr, 0, f1);
    fp_stage(xyz,  xyz1, nullptr, 0, f1, 128, 2048, 1024, F(37), 128, F(40), 128, F(43), 128, f0);

    // ---- heads ----
    float* out_seg  = (float*)d_out;                   // 16*2048*2
    float* out_cls  = out_seg + (size_t)NB * 2048 * 2; // 16*1024*40
    float* out_pred = out_cls + (size_t)NB * 1024 * 40;// 16*6

    conv_bn(F(64), 128, 128, f0, dbuf, 2048);                  // fc -> d

    conv_bn(F(67), 128, 128, dbuf, pA, 2048);                  // iou 1
    conv_bn(F(70), 64, 128, pA, pB, 2048);                     // iou 2
    conv(F(73), 3, 64, pB, pA, 2048, F(74), 0, 3);             // iou 3 (+bias)
    k_pred_s<<<1, 96, 0, stream>>>(pA, out_pred);

    // seg logits into 4-row-stride, pre-zeroed buffer (pad rows -> Kp=4 input)
    k_zero<<<gs_blocks((size_t)NB * 4 * 2048), 256, 0, stream>>>(pB, (size_t)NB * 4 * 2048);
    conv(F(75), 2, 128, dbuf, pB, 2048, F(76), 0, 4);          // seg logits (B,4(2),N)
    k_softmax2<<<ceil_div(NB * 2048, 256), 256, 0, stream>>>(pB, out_seg, 2048, 4);

    conv(F(77), 64, 2, pB, pA, 2048, F(78), 1, 64);            // feat 1 + relu (Kp=4)
    conv(F(79), 128, 64, pA, dbuf, 2048, F(80), 1, 128);       // feat 2 + relu
    conv(F(81), 40, 128, dbuf, pB, 2048, F(82), 0, 40);        // feat 3 (+bias)
    k_gather_cls<<<gs_blocks((size_t)NB * 1024 * 40), 256, 0, stream>>>(
        pB, perm, out_cls, 2048, 1024, 40);
}